// GraphTrainNN_71305047048897
// MI455X (gfx1250) — compile-verified
//
#include <hip/hip_runtime.h>
#include <hip/hip_bf16.h>
#include <stdint.h>

// Problem constants (match the reference).
#define B_ROWS   524288
#define G        127
#define ROWS     64              // rows per block tile == threads per block (1 thread : 1 row)
#define TPB      64              // 2 wave32 per block
#define N4       (ROWS * G / 4)  // 2032 16-byte chunks per tile (contiguous global span)

typedef __attribute__((__vector_size__(4 * sizeof(int)))) int i32x4;
typedef __attribute__((address_space(1))) i32x4 gbl_v4_t;   // global int4
typedef __attribute__((address_space(3))) i32x4 lds_v4_t;   // LDS int4

// 16 bytes, global -> LDS, via the gfx1250 async DMA path (tracked by ASYNCcnt).
__device__ __forceinline__ void async_b128_to_lds(const float* gsrc, float* lds_generic) {
#if __has_builtin(__builtin_amdgcn_global_load_async_to_lds_b128)
    __builtin_amdgcn_global_load_async_to_lds_b128(
        (gbl_v4_t*)(uintptr_t)gsrc,
        (lds_v4_t*)(uintptr_t)lds_generic,
        /*imm offset*/0, /*cpol*/0);
#else
    uint32_t laddr = (uint32_t)(uintptr_t)lds_generic;
    asm volatile("global_load_async_to_lds_b128 %0, %1, off"
                 :: "v"(laddr), "v"((uint64_t)(uintptr_t)gsrc)
                 : "memory");
#endif
}

__device__ __forceinline__ void wait_async_zero() {
#if __has_builtin(__builtin_amdgcn_s_wait_asynccnt)
    __builtin_amdgcn_s_wait_asynccnt(0);
#else
    asm volatile("s_wait_asynccnt 0" ::: "memory");
#endif
}

// h_g(x) = w2*(w1^2 + w0*x^2)/(w1^2 + x^2); out[b] = prod_g h_g(x[b,g])
// Restructured: out = prod[w2*(wn + w0*x^2)] / prod[wn + x^2]  -> ONE divide per row.
// LDS tile stride = 127 dwords; 127 = -1 mod 64, so lane t / column c reads bank
// (c - t) mod 64 -> conflict-free with NO padding -> tile is a verbatim copy of the
// contiguous global span -> async b128 DMA with zero address math.
__global__ __launch_bounds__(TPB) void hill_prod_kernel(const float* __restrict__ x,
                                                        const float* __restrict__ w,
                                                        float* __restrict__ out)
{
    __shared__ __align__(16) float tile[ROWS * G];   // 32512 B
    __shared__ __align__(16) float wpack[G * 4];     // {A=w2*w0, B=w2*wn, N=wn, 0} per gene

    const int tid = threadIdx.x;
    const long long base = (long long)blockIdx.x * (ROWS * G);  // dword index, 16B-aligned span

    // (1) Async-DMA the 64x127 tile (32512 contiguous bytes) into LDS, b128 chunks.
    for (int i4 = tid; i4 < N4; i4 += TPB)
        async_b128_to_lds(x + base + 4 * i4, &tile[4 * i4]);

    // (2) While the DMA is in flight, build packed per-gene constants (one 16B record each).
    for (int g = tid; g < G; g += TPB) {
        const float w0 = w[3 * g + 0];
        const float w1 = w[3 * g + 1];
        const float w2 = w[3 * g + 2];
        const float wn = w1 * w1;                 // N_EXP == 2
        wpack[4 * g + 0] = w2 * w0;
        wpack[4 * g + 1] = w2 * wn;
        wpack[4 * g + 2] = wn;
        wpack[4 * g + 3] = 0.0f;
    }

    // (3) Drain this wave's ASYNCcnt, then barrier so both waves see the tile.
    wait_async_zero();
    __syncthreads();

    // (4) One thread per row: 4 independent accumulator pairs for ILP;
    //     per column: 1 broadcast ds_load_b128 (weights) + 1 conflict-free ds_load (x).
    const float* trow = &tile[tid * G];
    float n0 = 1.0f, n1 = 1.0f, n2 = 1.0f, n3 = 1.0f;
    float d0 = 1.0f, d1 = 1.0f, d2 = 1.0f, d3 = 1.0f;

    int c = 0;
#pragma unroll 4
    for (; c + 4 <= G; c += 4) {
        const float4 w0v = *(const float4*)&wpack[4 * (c + 0)];
        const float4 w1v = *(const float4*)&wpack[4 * (c + 1)];
        const float4 w2v = *(const float4*)&wpack[4 * (c + 2)];
        const float4 w3v = *(const float4*)&wpack[4 * (c + 3)];
        const float x0 = trow[c + 0], x1 = trow[c + 1], x2 = trow[c + 2], x3 = trow[c + 3];
        const float s0 = x0 * x0, s1 = x1 * x1, s2 = x2 * x2, s3 = x3 * x3;
        n0 *= fmaf(w0v.x, s0, w0v.y);  d0 *= (w0v.z + s0);
        n1 *= fmaf(w1v.x, s1, w1v.y);  d1 *= (w1v.z + s1);
        n2 *= fmaf(w2v.x, s2, w2v.y);  d2 *= (w2v.z + s2);
        n3 *= fmaf(w3v.x, s3, w3v.y);  d3 *= (w3v.z + s3);
    }
    // Tail: columns 124..126.
    for (; c < G; ++c) {
        const float4 wv = *(const float4*)&wpack[4 * c];
        const float xv = trow[c];
        const float xn = xv * xv;
        n0 *= fmaf(wv.x, xn, wv.y);
        d0 *= (wv.z + xn);
    }

    const float num = (n0 * n1) * (n2 * n3);
    const float den = (d0 * d1) * (d2 * d3);
    out[(long long)blockIdx.x * ROWS + tid] = num / den;   // one precise divide per row
}

extern "C" void kernel_launch(void* const* d_in, const int* in_sizes, int n_in,
                              void* d_out, int out_size, void* d_ws, size_t ws_size,
                              hipStream_t stream) {
    const float* x = (const float*)d_in[0];   // (B, G) fp32
    const float* w = (const float*)d_in[1];   // (G, 3) fp32
    float* out = (float*)d_out;               // (B, 1) fp32
    const int nblocks = B_ROWS / ROWS;        // 8192
    hipLaunchKernelGGL(hill_prod_kernel, dim3(nblocks), dim3(TPB), 0, stream, x, w, out);
}